// RelationalGNN_48120813584781
// MI455X (gfx1250) — compile-verified
//
#include <hip/hip_runtime.h>
#include <stdint.h>

// Relational GNN message passing on MI455X (gfx1250).
//   per layer:  msg_e = h[src_e] * rel_emb[layer, etype_e]  (elementwise, D=64)
//               out   = segment_sum(msg, dst)               (atomic scatter)
// Bandwidth-bound (AI ~0.24 FLOP/B) -> optimize data movement, not matrix ops:
//  - rel table (64x64 f32 = 16KB) staged to LDS via CDNA5 async global->LDS DMA
//  - b128 gathers of source rows (node table resident in 192MB L2)
//  - hardware global_atomic_add_f32 (no-return) for the segment sum
//  - global_prefetch_b8 look-ahead on next gather row

#define D_FEAT 64
#define N_REL_MAX 64

__global__ __launch_bounds__(256) void rgnn_layer_kernel(
    const float* __restrict__ h_in,     // [n_nodes, 64]
    const float* __restrict__ rel,      // [n_rel, 64] (this layer's slice)
    const int*   __restrict__ src,      // [n_edges]
    const int*   __restrict__ dst,      // [n_edges]
    const int*   __restrict__ etype,    // [n_edges]
    float*       __restrict__ out,      // [n_nodes, 64], pre-zeroed
    int n_edges, int n_rel)
{
    __shared__ float s_rel[N_REL_MAX * D_FEAT];   // 16 KB of the 320 KB WGP LDS

    // --- Stage relation table into LDS with gfx1250 async global->LDS copies.
    // Each lane issues a 16B transfer from its global address to its LDS offset.
    {
        const int nf4 = (n_rel * D_FEAT) >> 2;            // float4 count (1024)
        for (int idx = threadIdx.x; idx < nf4; idx += blockDim.x) {
            uint32_t lds_off = (uint32_t)(uintptr_t)(&s_rel[idx * 4]); // low 32b of flat = LDS offset
            uint64_t gaddr   = (uint64_t)(uintptr_t)(rel + idx * 4);
            asm volatile("global_load_async_to_lds_b128 %0, %1, off"
                         :
                         : "v"(lds_off), "v"(gaddr)
                         : "memory");
        }
        asm volatile("s_wait_asynccnt 0" ::: "memory");   // drain ASYNCcnt before LDS reads
    }
    __syncthreads();

    // --- 16 lanes cooperate on one edge (64 floats = 16 x float4).
    const int lane = threadIdx.x & 15;            // float4 slot within the row
    const int grp  = threadIdx.x >> 4;            // edge-group within block
    const int grps_per_blk = blockDim.x >> 4;     // 16 groups @ 256 threads
    const int step = (int)gridDim.x * grps_per_blk;

    for (int e = (int)blockIdx.x * grps_per_blk + grp; e < n_edges; e += step) {
        const int s = src[e];
        const int r = etype[e];
        const int d = dst[e];

        // Prefetch the source row this group needs next iteration (hides L2 gather latency).
        const int en = e + step;
        if (en < n_edges) {
            __builtin_prefetch(h_in + (size_t)src[en] * D_FEAT, 0, 1);
        }

        const float4 hv = *(const float4*)(h_in + (size_t)s * D_FEAT + lane * 4);
        const float4 wv = *(const float4*)(&s_rel[r * D_FEAT + lane * 4]);

        float* o = out + (size_t)d * D_FEAT + lane * 4;
        // No-return f32 atomics -> global_atomic_add_f32 handled by L2 atomic units.
        __hip_atomic_fetch_add(o + 0, hv.x * wv.x, __ATOMIC_RELAXED, __HIP_MEMORY_SCOPE_AGENT);
        __hip_atomic_fetch_add(o + 1, hv.y * wv.y, __ATOMIC_RELAXED, __HIP_MEMORY_SCOPE_AGENT);
        __hip_atomic_fetch_add(o + 2, hv.z * wv.z, __ATOMIC_RELAXED, __HIP_MEMORY_SCOPE_AGENT);
        __hip_atomic_fetch_add(o + 3, hv.w * wv.w, __ATOMIC_RELAXED, __HIP_MEMORY_SCOPE_AGENT);
    }
}

__global__ __launch_bounds__(256) void rgnn_zero_kernel(float* __restrict__ p, int n4) {
    int i = (int)blockIdx.x * (int)blockDim.x + (int)threadIdx.x;
    const int stride = (int)gridDim.x * (int)blockDim.x;
    const float4 z = make_float4(0.f, 0.f, 0.f, 0.f);
    for (; i < n4; i += stride) {
        ((float4*)p)[i] = z;
    }
}

extern "C" void kernel_launch(void* const* d_in, const int* in_sizes, int n_in,
                              void* d_out, int out_size, void* d_ws, size_t ws_size,
                              hipStream_t stream) {
    // Inputs (setup_inputs order): node_features f32 [N,64], rel_emb f32 [L,64,64],
    //                              src i32 [E], dst i32 [E], etype i32 [E]
    const float* node_features = (const float*)d_in[0];
    const float* rel_emb       = (const float*)d_in[1];
    const int*   src           = (const int*)d_in[2];
    const int*   dst           = (const int*)d_in[3];
    const int*   etype         = (const int*)d_in[4];

    const int n_nodes  = in_sizes[0] / D_FEAT;
    const int n_edges  = in_sizes[2];
    const int n_rel    = N_REL_MAX;
    const int n_layers = in_sizes[1] / (n_rel * D_FEAT);

    float* ws   = (float*)d_ws;    // intermediate h: n_nodes*64 f32 = 12.8 MB (< ws_size)
    float* outp = (float*)d_out;

    const int nfeat4 = (n_nodes * D_FEAT) >> 2;    // 800K float4s = 12.8 MB
    const int zthreads = 256;
    int zblocks = (nfeat4 + zthreads - 1) / zthreads;
    if (zblocks > 8192) zblocks = 8192;            // grid-stride covers the rest

    const int eblocks = 4096;                      // persistent grid over 800K edges

    // Ping-pong so the final layer always writes d_out.
    const float* cur_in = node_features;
    for (int layer = 0; layer < n_layers; ++layer) {
        float* cur_out = (((n_layers - 1 - layer) & 1) == 0) ? outp : ws;

        rgnn_zero_kernel<<<zblocks, zthreads, 0, stream>>>(cur_out, nfeat4);

        rgnn_layer_kernel<<<eblocks, 256, 0, stream>>>(
            cur_in,
            rel_emb + (size_t)layer * n_rel * D_FEAT,
            src, dst, etype,
            cur_out, n_edges, n_rel);

        cur_in = cur_out;
    }
}